// VectorQuantizer_89867895701864
// MI455X (gfx1250) — compile-verified
//
#include <hip/hip_runtime.h>
#include <hip/hip_bf16.h>

// ---------------------------------------------------------------------------
// VQ-VAE vector quantizer forward for MI455X (gfx1250, wave32, WMMA).
// z: (32,4096,64) fp32 -> N=131072 rows, D=64.  codebook: (1024,64) fp32.
// Outputs (concatenated in d_out, all fp32):
//   [0, 8388608)            z_q (straight-through forward value = codebook[idx])
//   [8388608]               commitment loss = 1.25 * mean((z_q - z)^2)
//   [8388609, 8519681)      encoding indices as float, shape (32,4096)
//
// argmin_k( ||e_k||^2 - 2 z.e_k )  — the ||z||^2 term is row-constant and
// dropped.  Matmul z @ cb^T runs on V_WMMA_F32_16X16X4_F32 with the whole
// codebook staged in LDS (320KB/WGP on CDNA5), rows padded to 68 floats so
// B-fragment ds_load_b64 reads are bank-conflict-free across all 32 lanes.
// ---------------------------------------------------------------------------

#define VQ_D 64
#define VQ_K 1024
#define VQ_N 131072        // 32 * 4096
#define VQ_TOTAL (VQ_N * VQ_D)
#define CB_STRIDE 68       // padded LDS row stride (floats): conflict-free banks
#define LDS_FLOATS (VQ_K * CB_STRIDE + VQ_K)   // codebook + norms

typedef float v2f __attribute__((ext_vector_type(2)));
typedef float v8f __attribute__((ext_vector_type(8)));

// ---------------------------------------------------------------------------
// Kernel 1: codebook squared norms ||e_k||^2 -> cbn[k]; also zero the loss.
// ---------------------------------------------------------------------------
__global__ __launch_bounds__(256) void vq_norms_kernel(
    const float* __restrict__ cb, float* __restrict__ cbn,
    float* __restrict__ loss_out)
{
    int k = blockIdx.x * blockDim.x + threadIdx.x;
    if (k == 0) *loss_out = 0.0f;   // loss accumulator init (stream-ordered)
    if (k < VQ_K) {
        const float4* p = (const float4*)(cb + k * VQ_D);
        float s = 0.0f;
        #pragma unroll
        for (int j = 0; j < VQ_D / 4; ++j) {
            float4 v = p[j];
            s += v.x * v.x + v.y * v.y + v.z * v.z + v.w * v.w;
        }
        cbn[k] = s;
    }
}

// ---------------------------------------------------------------------------
// Kernel 2: fused distance + argmin.
// Block = 256 threads = 8 waves; each wave owns 32 z-rows (two 16-row tiles),
// so a block covers 256 rows.  Whole codebook lives in LDS.
// Per 16-column tile: 16 ds_load_b64 B fragments, each feeding 2 WMMAs
// (row-tile 0 and 1), split into 4 independent accumulator chains of depth 8.
// ---------------------------------------------------------------------------
__global__ __launch_bounds__(256) void vq_argmin_kernel(
    const float* __restrict__ z, const float* __restrict__ cb,
    const float* __restrict__ cbn,
    int* __restrict__ idx_ws, float* __restrict__ idx_out_f)
{
    extern __shared__ float lds[];
    float* lds_cb  = lds;                      // VQ_K rows x CB_STRIDE floats
    float* lds_cbn = lds + VQ_K * CB_STRIDE;   // VQ_K floats

    // ---- cooperative stage: codebook (256KB) + norms (4KB) into LDS ----
    {
        const float4* src = (const float4*)cb;   // 16 float4 per code row
        for (int i = threadIdx.x; i < VQ_K * 16; i += 256) {
            int k = i >> 4, j = i & 15;
            *(float4*)(lds_cb + k * CB_STRIDE + j * 4) = src[i];
        }
        for (int k = threadIdx.x; k < VQ_K; k += 256)
            lds_cbn[k] = cbn[k];
    }
    __syncthreads();

    const int lane = threadIdx.x & 31;
    const int wave = threadIdx.x >> 5;
    const int row0 = blockIdx.x * 256 + wave * 32;   // first of 32 rows

    const int m    = lane & 15;     // A-tile row / B-tile column for this lane
    const int half = lane >> 4;     // 0: K%4 in {0,1}, 1: K%4 in {2,3}
    const int kb   = half * 2;

    // ---- resident A fragments: two 16x64 strips (2 x 32 VGPRs) ----
    v2f a0[16], a1[16];
    {
        const float* zr0 = z + (size_t)(row0 + m) * VQ_D + kb;
        const float* zr1 = zr0 + 16 * VQ_D;
        #pragma unroll
        for (int j = 0; j < 16; ++j) {
            a0[j] = *(const v2f*)(zr0 + 4 * j);
            a1[j] = *(const v2f*)(zr1 + 4 * j);
        }
    }

    float bd0[8], bd1[8];
    int   bi0[8], bi1[8];
    #pragma unroll
    for (int r = 0; r < 8; ++r) {
        bd0[r] = 3.402823e38f; bi0[r] = 0;
        bd1[r] = 3.402823e38f; bi1[r] = 0;
    }

    for (int nb = 0; nb < VQ_K; nb += 16) {
        const float* br = lds_cb + (nb + m) * CB_STRIDE + kb;

        v8f c00 = {0.f,0.f,0.f,0.f,0.f,0.f,0.f,0.f};   // tile0, even j
        v8f c01 = {0.f,0.f,0.f,0.f,0.f,0.f,0.f,0.f};   // tile0, odd  j
        v8f c10 = {0.f,0.f,0.f,0.f,0.f,0.f,0.f,0.f};   // tile1, even j
        v8f c11 = {0.f,0.f,0.f,0.f,0.f,0.f,0.f,0.f};   // tile1, odd  j
        #pragma unroll
        for (int j = 0; j < 16; j += 2) {
            v2f b0 = *(const v2f*)(br + 4 * j);
            v2f b1 = *(const v2f*)(br + 4 * (j + 1));
            c00 = __builtin_amdgcn_wmma_f32_16x16x4_f32(false, a0[j],   false, b0, (short)0, c00, false, false);
            c10 = __builtin_amdgcn_wmma_f32_16x16x4_f32(false, a1[j],   false, b0, (short)0, c10, false, false);
            c01 = __builtin_amdgcn_wmma_f32_16x16x4_f32(false, a0[j+1], false, b1, (short)0, c01, false, false);
            c11 = __builtin_amdgcn_wmma_f32_16x16x4_f32(false, a1[j+1], false, b1, (short)0, c11, false, false);
        }

        const float nrm  = lds_cbn[nb + m];   // ||e_n||^2, n = this lane's column
        const int   cand = nb + m;
        #pragma unroll
        for (int r = 0; r < 8; ++r) {
            float d0 = __builtin_fmaf(-2.0f, c00[r] + c01[r], nrm);
            if (d0 < bd0[r]) { bd0[r] = d0; bi0[r] = cand; }
            float d1 = __builtin_fmaf(-2.0f, c10[r] + c11[r], nrm);
            if (d1 < bd1[r]) { bd1[r] = d1; bi1[r] = cand; }
        }
    }

    // ---- min+argmin across the 16 lanes of each half, then write ----
    #pragma unroll
    for (int t = 0; t < 2; ++t) {
        #pragma unroll
        for (int r = 0; r < 8; ++r) {
            float d = t ? bd1[r] : bd0[r];
            int   i = t ? bi1[r] : bi0[r];
            #pragma unroll
            for (int off = 8; off >= 1; off >>= 1) {
                float od = __shfl_xor(d, off, 16);
                int   oi = __shfl_xor(i, off, 16);
                if (od < d || (od == d && oi < i)) { d = od; i = oi; }
            }
            if (m == 0) {
                int row = row0 + t * 16 + r + half * 8;
                idx_ws[row]    = i;
                idx_out_f[row] = (float)i;
            }
        }
    }
}

// ---------------------------------------------------------------------------
// Kernel 3: gather z_q = codebook[idx], write out, accumulate
//           loss = 1.25/(N*D) * sum((z_q - z)^2)  via f32 global atomic.
// ---------------------------------------------------------------------------
__global__ __launch_bounds__(256) void vq_gather_loss_kernel(
    const float* __restrict__ z, const float* __restrict__ cb,
    const int* __restrict__ idx_ws,
    float* __restrict__ zq_out, float* __restrict__ loss)
{
    int t = blockIdx.x * 256 + threadIdx.x;
    float sq = 0.0f;
    if (t < VQ_TOTAL) {
        int row = t >> 6;        // / VQ_D
        int d   = t & 63;        // % VQ_D
        int i   = idx_ws[row];
        float q = cb[(size_t)i * VQ_D + d];
        zq_out[t] = q;
        float diff = q - z[t];
        sq = diff * diff;
    }
    // wave32 reduction
    #pragma unroll
    for (int off = 16; off >= 1; off >>= 1)
        sq += __shfl_xor(sq, off, 32);

    __shared__ float smem[8];
    int lane = threadIdx.x & 31, w = threadIdx.x >> 5;
    if (lane == 0) smem[w] = sq;
    __syncthreads();
    if (w == 0) {
        float v = (lane < 8) ? smem[lane] : 0.0f;
        #pragma unroll
        for (int off = 4; off >= 1; off >>= 1)
            v += __shfl_xor(v, off, 8);
        if (lane == 0)
            atomicAdd(loss, v * (1.25f / (float)VQ_TOTAL));
    }
}

// ---------------------------------------------------------------------------
extern "C" void kernel_launch(void* const* d_in, const int* in_sizes, int n_in,
                              void* d_out, int out_size, void* d_ws, size_t ws_size,
                              hipStream_t stream) {
    const float* z  = (const float*)d_in[0];   // (32,4096,64) fp32
    const float* cb = (const float*)d_in[1];   // (1024,64) fp32

    float* out  = (float*)d_out;
    float* zq   = out;                 // 8388608 floats
    float* loss = out + VQ_TOTAL;      // 1 float
    float* idxf = out + VQ_TOTAL + 1;  // 131072 floats (indices as float)

    float* cbn    = (float*)d_ws;                       // 1024 floats
    int*   idx_ws = (int*)((char*)d_ws + 4096);         // 131072 int32

    const size_t lds_bytes = (size_t)LDS_FLOATS * sizeof(float);  // 282624 B

    vq_norms_kernel<<<(VQ_K + 255) / 256, 256, 0, stream>>>(cb, cbn, loss);
    vq_argmin_kernel<<<VQ_N / 256, 256, lds_bytes, stream>>>(z, cb, cbn, idx_ws, idxf);
    vq_gather_loss_kernel<<<VQ_TOTAL / 256, 256, 0, stream>>>(z, cb, idx_ws, zq, loss);
}